// MILR_42107859370851
// MI455X (gfx1250) — compile-verified
//
#include <hip/hip_runtime.h>
#include <hip/hip_bf16.h>
#include <math.h>

typedef float v2f __attribute__((ext_vector_type(2)));
typedef float v8f __attribute__((ext_vector_type(8)));

// ---------------------------------------------------------------------------
// Kernel 1: inst[i] = dot(X[i,:], W) + b  via V_WMMA_F32_16X16X4_F32.
// One wave computes 16 rows per tile: A = X tile [16 rows x 4 K], B = W
// replicated across all 16 columns, C accumulates across 8 WMMAs (K=32).
// Memory-bound: 256 MB of X at 23.3 TB/s ~ 11 us; WMMA cost is negligible.
// ---------------------------------------------------------------------------
__global__ void mil_inst_wmma(const float* __restrict__ X,
                              const float* __restrict__ W,
                              const float* __restrict__ bias_p,
                              float* __restrict__ inst,
                              int N)
{
    const int lane          = threadIdx.x & 31;
    const int wavesPerBlock = blockDim.x >> 5;
    const int waveId        = blockIdx.x * wavesPerBlock + (threadIdx.x >> 5);
    const int numWaves      = gridDim.x * wavesPerBlock;
    const int half          = lane >> 4;   // 0: lanes 0-15, 1: lanes 16-31
    const int m             = lane & 15;   // row within tile (A layout)
    const int khalf         = half * 2;    // K pair this half-wave supplies

    // B fragments (4x16, f32): lanes 0-15 carry K=0(K=4j+0) in .x, K=4j+1 in .y;
    // lanes 16-31 carry K=4j+2 / 4j+3. W replicated over all 16 N-columns, so
    // every lane's C VGPRs hold valid row results regardless of column index.
    v2f bfrag[8];
#pragma unroll
    for (int j = 0; j < 8; ++j) {
        bfrag[j].x = W[4 * j + khalf + 0];
        bfrag[j].y = W[4 * j + khalf + 1];
    }
    const float bias = bias_p[0];

    const int tiles = N >> 4;
    for (int tile = waveId; tile < tiles; tile += numWaves) {
        const int row0 = tile << 4;
        // A-fragment source: lane supplies row (row0+m), K pair (4j+khalf .. +1)
        const float* xrow = X + (size_t)(row0 + m) * 32 + khalf;
        v8f c = {0.f, 0.f, 0.f, 0.f, 0.f, 0.f, 0.f, 0.f};
#pragma unroll
        for (int j = 0; j < 8; ++j) {
            v2f a = *(const v2f*)(xrow + j * 4);   // 8B-aligned global_load_b64
            c = __builtin_amdgcn_wmma_f32_16x16x4_f32(
                    /*neg_a=*/false, a, /*neg_b=*/false, bfrag[j],
                    /*c_mod=*/(short)0, c, /*reuse_a=*/false, /*reuse_b=*/false);
        }
        // C/D layout: VGPR r, lanes 0-15 -> M=r, lanes 16-31 -> M=r+8.
        // All columns identical, so lane 0 stores rows 0-7, lane 16 rows 8-15.
        if (m == 0) {
            float* dst = inst + row0 + half * 8;
            float4 lo = make_float4(c[0] + bias, c[1] + bias, c[2] + bias, c[3] + bias);
            float4 hi = make_float4(c[4] + bias, c[5] + bias, c[6] + bias, c[7] + bias);
            *(float4*)(dst + 0) = lo;
            *(float4*)(dst + 4) = hi;
        }
    }

    // Remainder rows if N % 16 != 0 (not hit for N = 2,000,000; kept for safety).
    const int rem0 = tiles << 4;
    for (int r = rem0 + waveId * 32 + lane; r < N; r += numWaves * 32) {
        float acc = bias;
        const float* xr = X + (size_t)r * 32;
#pragma unroll
        for (int k = 0; k < 32; ++k) acc = fmaf(xr[k], W[k], acc);
        inst[r] = acc;
    }
}

// ---------------------------------------------------------------------------
// Kernel 2: per-bag masked logsumexp-mean + 2-class log-softmax.
// One wave32 per bag; 2 elements per lane; butterfly reductions.
// inst (8 MB) is L2-resident (192 MB L2), so gathers are L2 hits.
// ---------------------------------------------------------------------------
__device__ __forceinline__ float softplus_f(float x)
{
    // numerically stable softplus: max(x,0) + log1p(exp(-|x|))
    return fmaxf(x, 0.f) + log1pf(expf(-fabsf(x)));
}

__global__ void mil_bag_reduce(const float* __restrict__ inst,
                               const int* __restrict__ bags,
                               const unsigned char* __restrict__ mask,  // bool, 1B
                               const float* __restrict__ sparam,
                               float* __restrict__ out,
                               int B)
{
    const int lane = threadIdx.x & 31;
    const int bag  = (int)((blockIdx.x * blockDim.x + threadIdx.x) >> 5);
    if (bag >= B) return;

    const float sp = softplus_f(sparam[0]);

    const long long base = (long long)bag * 64;
    const int  i0 = bags[base + lane];
    const int  i1 = bags[base + 32 + lane];
    const bool v0 = (mask[base + lane] == 0);
    const bool v1 = (mask[base + 32 + lane] == 0);

    const float NEG_INF = -__builtin_inff();
    const float x0 = v0 ? inst[i0] * sp : NEG_INF;
    const float x1 = v1 ? inst[i1] * sp : NEG_INF;

    float mx = fmaxf(x0, x1);
#pragma unroll
    for (int off = 16; off > 0; off >>= 1)
        mx = fmaxf(mx, __shfl_xor(mx, off, 32));

    float s  = (v0 ? expf(x0 - mx) : 0.f) + (v1 ? expf(x1 - mx) : 0.f);
    float cf = (v0 ? 1.f : 0.f) + (v1 ? 1.f : 0.f);   // unmasked count
#pragma unroll
    for (int off = 16; off > 0; off >>= 1) {
        s  += __shfl_xor(s,  off, 32);
        cf += __shfl_xor(cf, off, 32);
    }

    if (lane == 0) {
        const float z   = (mx + logf(s / cf)) / sp;   // bag logit
        const float spz = softplus_f(z);              // log(1 + e^z)
        out[2 * bag + 0] = -spz;                      // log_softmax class 0
        out[2 * bag + 1] = z - spz;                   // log_softmax class 1
    }
}

// ---------------------------------------------------------------------------
// setup_inputs order: X[N*32] f32, W[32] f32, b[1] f32, softmax_parameter[1]
// f32, bags[B*64] i32, padding_mask[B*64] bool(1B). Output: [B,2] f32.
// ---------------------------------------------------------------------------
extern "C" void kernel_launch(void* const* d_in, const int* in_sizes, int n_in,
                              void* d_out, int out_size, void* d_ws, size_t ws_size,
                              hipStream_t stream)
{
    const float*         X      = (const float*)d_in[0];
    const float*         W      = (const float*)d_in[1];
    const float*         b      = (const float*)d_in[2];
    const float*         sparam = (const float*)d_in[3];
    const int*           bags   = (const int*)d_in[4];
    const unsigned char* mask   = (const unsigned char*)d_in[5];
    float*               out    = (float*)d_out;

    const int N = in_sizes[0] / 32;   // 2,000,000
    const int B = in_sizes[4] / 64;   // 50,000

    float* inst = (float*)d_ws;       // N floats = 8 MB scratch

    const int threads       = 256;    // 8 wave32s per block
    const int wavesPerBlock = threads / 32;

    int tiles   = N / 16;
    int blocks1 = (tiles + wavesPerBlock - 1) / wavesPerBlock;
    if (blocks1 > 4096) blocks1 = 4096;   // ~32k waves, each loops over tiles
    if (blocks1 < 1)    blocks1 = 1;
    mil_inst_wmma<<<blocks1, threads, 0, stream>>>(X, W, b, inst, N);

    int blocks2 = (B + wavesPerBlock - 1) / wavesPerBlock;
    mil_bag_reduce<<<blocks2, threads, 0, stream>>>(inst, bags, mask, sparam, out, B);
}